// LstmEncoder_40604620816711
// MI455X (gfx1250) — compile-verified
//
#include <hip/hip_runtime.h>
#include <math.h>
#include <stdint.h>

typedef __attribute__((ext_vector_type(2))) float v2f;
typedef __attribute__((ext_vector_type(8))) float v8f;
typedef __attribute__((ext_vector_type(4))) unsigned int v4u;
typedef __attribute__((ext_vector_type(4))) int          v4i;
typedef __attribute__((ext_vector_type(8))) int          v8i;

#define Bv   64
#define Tv   512
#define Ev   256
#define Uv   256
#define Gv   1024          // 4*U
#define NBLK 16            // workgroups in the recurrence grid

__device__ __forceinline__ float sigm(float x) { return 1.0f / (1.0f + __expf(-x)); }

__global__ void zero_counters_kernel(int* __restrict__ cnt, int n) {
  int i = blockIdx.x * blockDim.x + threadIdx.x;
  if (i < n) cnt[i] = 0;
}

// ---------------- TDM descriptor packing (ISA cdna5 ch.8) ------------------
// Group0: [1:0]=count=1, [63:32]=lds byte addr, [120:64]=global addr, [127:126]=type=2
__device__ __forceinline__ v4u tdm_g0(unsigned lds_addr, unsigned long long ga) {
  v4u g;
  g.x = 1u;
  g.y = lds_addr;
  g.z = (unsigned)(ga & 0xffffffffull);
  g.w = (unsigned)((ga >> 32) & 0x1ffffffull) | (2u << 30);
  return g;
}
// Group1: data_size=2 (4B), tensor_dim0@[79:48], tensor_dim1@[111:80],
// tile_dim0@[127:112], tile_dim1@[143:128], tile_dim2=0,
// tensor_dim0_stride@[207:160], tensor_dim1_stride@[255:208]
__device__ __forceinline__ v8i tdm_g1(unsigned td0, unsigned td1,
                                      unsigned tile0, unsigned tile1,
                                      unsigned long long s0, unsigned long long s1) {
  v8i g;
  g[0] = 0x00020000;                                   // data_size=4B; no flags
  g[1] = (int)((td0 & 0xffffu) << 16);                 // abar_addr=0 | td0.lo16
  g[2] = (int)((td0 >> 16) | ((td1 & 0xffffu) << 16));
  g[3] = (int)((td1 >> 16) | ((tile0 & 0xffffu) << 16));
  g[4] = (int)(tile1 & 0xffffu);                       // tile_dim1 | tile_dim2=0
  g[5] = (int)(s0 & 0xffffffffull);
  g[6] = (int)((s0 >> 32) & 0xffffull) | (int)((s1 & 0xffffull) << 16);
  g[7] = (int)((s1 >> 16) & 0xffffffffull);
  return g;
}

// ---------------------------------------------------------------------------
// out[M,1024] = A[M,256] @ W[256,1024] + bias[1024]; 2Mx2N register blocking:
// each wave computes a 32x32 block as 4 WMMA tiles, reusing A/B fragments 2x.
__global__ __launch_bounds__(256) void gemm_xz_kernel(
    const int*   __restrict__ tokens,
    const float* __restrict__ emb,
    const float* __restrict__ Adense,
    const float* __restrict__ W,
    const float* __restrict__ bias,
    float*       __restrict__ out,
    int M)
{
  const int lane = threadIdx.x & 31;
  const int wid  = blockIdx.x * (blockDim.x >> 5) + (threadIdx.x >> 5);
  const int nt2  = wid & 31;          // 1024/32 = 32 N blocks
  const int mt2  = wid >> 5;
  if (mt2 * 32 >= M) return;

  const int half = lane >> 4;
  const int l15  = lane & 15;
  const int arow0 = mt2 * 32 + l15, arow1 = arow0 + 16;
  const int bcol0 = nt2 * 32 + l15, bcol1 = bcol0 + 16;

  const float* A0 = tokens ? (emb + (size_t)tokens[arow0] * Ev)
                           : (Adense + (size_t)arow0 * Ev);
  const float* A1 = tokens ? (emb + (size_t)tokens[arow1] * Ev)
                           : (Adense + (size_t)arow1 * Ev);

  v8f a00 = {}, a01 = {}, a10 = {}, a11 = {};
  for (int kk = 0; kk < Ev / 4; ++kk) {
    const int k = kk * 4 + half * 2;
    v2f fa0, fa1, fb0, fb1;
    fa0.x = A0[k];     fa0.y = A0[k + 1];
    fa1.x = A1[k];     fa1.y = A1[k + 1];
    fb0.x = W[(size_t)k * Gv + bcol0];  fb0.y = W[(size_t)(k + 1) * Gv + bcol0];
    fb1.x = W[(size_t)k * Gv + bcol1];  fb1.y = W[(size_t)(k + 1) * Gv + bcol1];
    a00 = __builtin_amdgcn_wmma_f32_16x16x4_f32(false, fa0, false, fb0, (short)0, a00, false, false);
    a01 = __builtin_amdgcn_wmma_f32_16x16x4_f32(false, fa0, false, fb1, (short)0, a01, false, false);
    a10 = __builtin_amdgcn_wmma_f32_16x16x4_f32(false, fa1, false, fb0, (short)0, a10, false, false);
    a11 = __builtin_amdgcn_wmma_f32_16x16x4_f32(false, fa1, false, fb1, (short)0, a11, false, false);
  }

  const float bs0 = bias[bcol0], bs1 = bias[bcol1];
#pragma unroll
  for (int j = 0; j < 8; ++j) {
    const int r0 = mt2 * 32 + j + half * 8;       // C layout: VGPR j -> row j (+8 hi)
    out[(size_t)r0 * Gv + nt2 * 32 + l15]        = a00[j] + bs0;
    out[(size_t)r0 * Gv + nt2 * 32 + 16 + l15]   = a01[j] + bs1;
    out[(size_t)(r0 + 16) * Gv + nt2 * 32 + l15]      = a10[j] + bs0;
    out[(size_t)(r0 + 16) * Gv + nt2 * 32 + 16 + l15] = a11[j] + bs1;
  }
}

__device__ __forceinline__ void grid_barrier(int* cnt, int slot) {
  __syncthreads();
  __builtin_amdgcn_s_cluster_barrier();   // NOP unless dispatched as a cluster
  if (threadIdx.x == 0) {
    __threadfence();
    __hip_atomic_fetch_add(cnt + slot, 1, __ATOMIC_RELEASE, __HIP_MEMORY_SCOPE_AGENT);
    while (__hip_atomic_load(cnt + slot, __ATOMIC_ACQUIRE, __HIP_MEMORY_SCOPE_AGENT) < NBLK) {
      __builtin_amdgcn_s_sleep(1);
    }
  }
  __syncthreads();
}

// ---------------------------------------------------------------------------
// Masked-LSTM recurrence. Block bid owns units [bid*16, bid*16+16) (all 4 gates).
// R column-slice is TDM-staged to LDS once; h_prev TDM-staged each step.
// Dynamic LDS: hlds 64KB | rlds 64KB (4 panels [256][16]) | zbuf 16KB | cs 4KB.
__global__ __launch_bounds__(256) void lstm_rec_kernel(
    const int*   __restrict__ tokens,   // [B,T]
    const float* __restrict__ xz,       // [B,T,1024]
    const float* __restrict__ R,        // [256,1024]
    float*       __restrict__ outseq,   // [B,T,256]
    float*       __restrict__ hbuf,     // [2,64,256] ping-pong h
    float*       __restrict__ hfinal,   // [64,256] or nullptr
    int*         __restrict__ barcnt,
    int slotbase)
{
  extern __shared__ float smem[];
  float* hlds = smem;                       // [64][256]
  float* rlds = hlds + Bv * Uv;             // 4 * [256][16]
  float* zbuf = rlds + 4 * Uv * 16;         // [64][64]
  float* cs   = zbuf + 64 * 64;             // [64][16]

  const int tid  = threadIdx.x;
  const int bid  = blockIdx.x;              // 0..15
  const int lane = tid & 31;
  const int wv   = tid >> 5;                // 0..7
  const int half = lane >> 4;
  const int l15  = lane & 15;
  const int ub   = bid * 16;

  // init: c = 0; zero owned slice of both h ping-pong buffers
  for (int i = tid; i < 64 * 16; i += 256) cs[i] = 0.0f;
  for (int i = tid; i < 2 * 64 * 16; i += 256) {
    const int buf = i >> 10, row = (i >> 4) & 63, j = i & 15;
    hbuf[(size_t)buf * Bv * Uv + row * Uv + ub + j] = 0.0f;
  }

  // TDM: stage the 4 gate panels of R (16 cols x 256 K each) into LDS once.
  const v4i z4 = {};
  const v8i z8 = {};
  if (wv == 0) {
#pragma unroll
    for (int g = 0; g < 4; ++g) {
      const unsigned lds_a = (unsigned)(uintptr_t)(rlds + g * Uv * 16);
      const v4u g0 = tdm_g0(lds_a, (unsigned long long)(uintptr_t)(R + g * Uv + ub));
      const v8i g1 = tdm_g1(/*td0*/16, /*td1*/Uv, /*tile0*/16, /*tile1*/Uv,
                            /*s0*/Gv, /*s1*/0);
      __builtin_amdgcn_tensor_load_to_lds(g0, g1, z4, z4, z8, 0);
    }
  }
  grid_barrier(barcnt, slotbase);           // initial h zeros visible everywhere

  const v8i g1h = tdm_g1(/*td0*/Bv * Uv, /*td1*/1, /*tile0*/Bv * Uv, /*tile1*/0,
                         /*s0*/Bv * Uv, /*s1*/0);

  for (int t = 0; t < Tv; ++t) {
    const float* hprev = hbuf + (size_t)((t + 1) & 1) * Bv * Uv;
    float*       hnext = hbuf + (size_t)(t & 1) * Bv * Uv;

    // TDM: stage h_prev (64x256 f32 = 64KB, contiguous) into LDS.
    if (wv == 0) {
      const v4u g0 = tdm_g0((unsigned)(uintptr_t)hlds,
                            (unsigned long long)(uintptr_t)hprev);
      __builtin_amdgcn_tensor_load_to_lds(g0, g1h, z4, z4, z8, 0);
      __builtin_amdgcn_s_wait_tensorcnt(0);   // also covers the R panels (in-order)
    }
    __syncthreads();

    // z-slice = hlds @ rlds ; 16 tiles (4 batch x 4 gates), 2 per wave
#pragma unroll
    for (int tt = 0; tt < 2; ++tt) {
      const int tile = wv + tt * 8;
      const int mt = tile >> 2;             // batch tile 0..3
      const int ct = tile & 3;              // gate 0..3
      const int arow = mt * 16 + l15;
      const float* rp = rlds + ct * Uv * 16 + l15;
      v8f acc = {};
      for (int kk = 0; kk < Uv / 4; ++kk) {
        const int k = kk * 4 + half * 2;
        v2f a, b;
        a.x = hlds[arow * Uv + k];
        a.y = hlds[arow * Uv + k + 1];
        b.x = rp[k * 16];
        b.y = rp[(k + 1) * 16];
        acc = __builtin_amdgcn_wmma_f32_16x16x4_f32(false, a, false, b,
                                                    (short)0, acc, false, false);
      }
#pragma unroll
      for (int j = 0; j < 8; ++j)
        zbuf[(mt * 16 + j + half * 8) * 64 + ct * 16 + l15] = acc[j];
    }
    __syncthreads();

    // gates + masked state update for 64 rows x 16 owned units
    for (int e = tid; e < Bv * 16; e += 256) {
      const int row = e >> 4;
      const int j   = e & 15;
      const float* xzp = xz + ((size_t)row * Tv + t) * Gv + ub + j;
      __builtin_prefetch(xzp + Gv, 0, 0);   // next-step xz -> global_prefetch_b8

      const float zi = zbuf[row * 64 +  0 + j] + xzp[0 * Uv];
      const float zf = zbuf[row * 64 + 16 + j] + xzp[1 * Uv];
      const float zg = zbuf[row * 64 + 32 + j] + xzp[2 * Uv];
      const float zo = zbuf[row * 64 + 48 + j] + xzp[3 * Uv];

      const float c0 = cs[e];
      const float ig = sigm(zi);
      const float fg = sigm(zf);
      const float og = sigm(zo);
      const float cn = fg * c0 + ig * tanhf(zg);
      const float hn = og * tanhf(cn);

      const bool  m  = (tokens[row * Tv + t] == 0);   // mask = (x==0), True = valid
      const float hp = hlds[row * Uv + ub + j];
      const float h2 = m ? hn : hp;
      cs[e] = m ? cn : c0;

      hnext[row * Uv + ub + j] = h2;
      outseq[((size_t)row * Tv + t) * Uv + ub + j] = h2;
      if (hfinal && t == Tv - 1) hfinal[row * Uv + ub + j] = h2;
    }
    grid_barrier(barcnt, slotbase + 1 + t);   // publish h before anyone reads it
  }
}

extern "C" void kernel_launch(void* const* d_in, const int* in_sizes, int n_in,
                              void* d_out, int out_size, void* d_ws, size_t ws_size,
                              hipStream_t stream)
{
  const int*   x   = (const int*)  d_in[0];
  const float* emb = (const float*)d_in[1];
  const float* k0  = (const float*)d_in[2];
  const float* r0  = (const float*)d_in[3];
  const float* b0  = (const float*)d_in[4];
  const float* k1  = (const float*)d_in[5];
  const float* r1  = (const float*)d_in[6];
  const float* b1  = (const float*)d_in[7];
  float* out = (float*)d_out;                 // [B,T,U] then final h [B,U]

  char* ws = (char*)d_ws;
  float* xz   = (float*)ws;                                       // B*T*1024 f32
  size_t off  = (size_t)Bv * Tv * Gv * sizeof(float);
  float* out0 = (float*)(ws + off);                               // B*T*256 f32
  off += (size_t)Bv * Tv * Uv * sizeof(float);
  float* hbuf = (float*)(ws + off);                               // 2*64*256 f32
  off += (size_t)2 * Bv * Uv * sizeof(float);
  int*   cnt  = (int*)(ws + off);                                 // barrier counters

  const int M      = Bv * Tv;                    // 32768
  const int nslots = 2 * (Tv + 1);               // 1026
  const int blk32  = (M / 32) * (Gv / 32) / 8;   // waves/8 per 256-thread block
  const size_t smem = (size_t)(Bv * Uv + 4 * Uv * 16 + 64 * 64 + 64 * 16) * sizeof(float);

  zero_counters_kernel<<<(nslots + 255) / 256, 256, 0, stream>>>(cnt, nslots);

  // layer 0: xz = emb[x] @ k0 + b0   (gather GEMM)
  gemm_xz_kernel<<<blk32, 256, 0, stream>>>(x, emb, (const float*)nullptr,
                                            k0, b0, xz, M);
  lstm_rec_kernel<<<NBLK, 256, smem, stream>>>(x, xz, r0, out0, hbuf,
                                               (float*)nullptr, cnt, 0);

  // layer 1: xz = out0 @ k1 + b1     (dense GEMM, reuse xz buffer)
  gemm_xz_kernel<<<blk32, 256, 0, stream>>>((const int*)nullptr,
                                            (const float*)nullptr, out0,
                                            k1, b1, xz, M);
  lstm_rec_kernel<<<NBLK, 256, smem, stream>>>(x, xz, r1, out, hbuf,
                                               out + (size_t)Bv * Tv * Uv, cnt, Tv + 1);
}